// BAddBMM_30700426232148
// MI455X (gfx1250) — compile-verified
//
#include <hip/hip_runtime.h>
#include <hip/hip_bf16.h>

// ---------------------------------------------------------------------------
// y[n,i] = relu( b[idx[n],i] + sum_o w[idx[n],i,o] * x[n,o] )
// BATCH=8192, N_MODELS=64, F=256 (in==out==256), all fp32.
//
// Strategy: counting-sort samples by model id, then per-model tiled GEMM on
// the fp32 WMMA pipe (V_WMMA_F32_16X16X4_F32), 16x16 output tiles per wave32.
// ---------------------------------------------------------------------------

#define N_MODELS 64
#define FDIM     256
#define TILE_M   16
#define XT_STRIDE 260   // 256 + 4 pad floats -> conflict-free LDS reads, 16B aligned rows

typedef float v2f __attribute__((ext_vector_type(2)));
typedef float v8f __attribute__((ext_vector_type(8)));

// ---------------- sorting kernels ----------------

__global__ void zero_counts_kernel(int* counts) {
    if (threadIdx.x < N_MODELS) counts[threadIdx.x] = 0;
}

__global__ void count_kernel(const int* __restrict__ idxs, int* __restrict__ counts, int n) {
    int i = blockIdx.x * blockDim.x + threadIdx.x;
    if (i < n) atomicAdd(&counts[idxs[i]], 1);
}

__global__ void scan_kernel(const int* __restrict__ counts, int* __restrict__ offsets,
                            int* __restrict__ cursor) {
    if (threadIdx.x == 0) {
        int acc = 0;
        for (int i = 0; i < N_MODELS; ++i) {
            offsets[i] = acc;
            cursor[i]  = acc;
            acc += counts[i];
        }
    }
}

__global__ void scatter_kernel(const int* __restrict__ idxs, int* __restrict__ cursor,
                               int* __restrict__ perm, int n) {
    int i = blockIdx.x * blockDim.x + threadIdx.x;
    if (i < n) {
        int p = atomicAdd(&cursor[idxs[i]], 1);
        perm[p] = i;
    }
}

// ---------------- grouped GEMM kernel ----------------
// grid: (maxRowTiles, N_MODELS); block: 512 threads = 16 wave32s.
// Block handles 16 samples (one row tile of model g) x all 256 outputs.
// Wave `w` handles output columns [16w, 16w+16).

__global__ __launch_bounds__(512)
void moe_wmma_gemm_kernel(const float* __restrict__ x,
                          const float* __restrict__ w,
                          const float* __restrict__ bias,
                          const int* __restrict__ counts,
                          const int* __restrict__ offsets,
                          const int* __restrict__ perm,
                          float* __restrict__ out) {
    const int g    = blockIdx.y;
    const int tile = blockIdx.x;
    const int count = counts[g];
    if (tile * TILE_M >= count) return;           // block-uniform early exit
    const int off = offsets[g];

    __shared__ float xtile[TILE_M * XT_STRIDE];
    __shared__ int   s_samp[TILE_M];

    // ---- stage gathered X tile (16 rows x 256 f32) into LDS ----
    // 16 rows * 64 float4 chunks = 1024 chunks; 512 threads -> 2 chunks each.
    for (int c = threadIdx.x; c < TILE_M * 64; c += blockDim.x) {
        const int r  = c >> 6;
        const int cv = (c & 63) << 2;
        const int rlocal = tile * TILE_M + r;
        float4 v = make_float4(0.f, 0.f, 0.f, 0.f);
        if (rlocal < count) {
            const int s = perm[off + rlocal];
            v = *(const float4*)(x + (size_t)s * FDIM + cv);
        }
        *(float4*)(&xtile[r * XT_STRIDE + cv]) = v;
    }
    if (threadIdx.x < TILE_M) {
        const int rlocal = tile * TILE_M + threadIdx.x;
        s_samp[threadIdx.x] = (rlocal < count) ? perm[off + rlocal] : -1;
    }
    __syncthreads();

    const int lane = threadIdx.x & 31;
    const int wave = threadIdx.x >> 5;     // 0..15 -> output column tile
    const int half = lane >> 4;            // 0: K=k0,k0+1 ; 1: K=k0+2,k0+3
    const int nn   = lane & 15;            // M for A-frag, N for B/C/D frags
    const int i0   = wave * 16;

    // C init: bias broadcast down M (same value in all 8 acc VGPRs per lane)
    const float bb = bias[g * FDIM + i0 + nn];
    v8f acc = {bb, bb, bb, bb, bb, bb, bb, bb};

    // A from LDS: A[m][k] = xtile[m][k];   lane -> (m=nn, k=k0+2*half+{0,1})
    // B from global: B[k][n] = w[g][i0+n][k]; lane -> (n=nn, k=k0+2*half+{0,1})
    const float* xrow = &xtile[nn * XT_STRIDE + half * 2];
    const float* wrow = w + (((size_t)g * FDIM) + i0 + nn) * FDIM + half * 2;

#pragma unroll 8
    for (int k0 = 0; k0 < FDIM; k0 += 4) {
        v2f a  = *(const v2f*)(xrow + k0);
        v2f bf = *(const v2f*)(wrow + k0);
        acc = __builtin_amdgcn_wmma_f32_16x16x4_f32(
            /*neg_a=*/false, a, /*neg_b=*/false, bf,
            /*c_mod=*/(short)0, acc, /*reuse_a=*/false, /*reuse_b=*/false);
    }

    // D layout: VGPR j -> rows M=j (lanes 0-15) and M=j+8 (lanes 16-31), N=nn
    float* ocol = out + i0 + nn;
#pragma unroll
    for (int j = 0; j < 8; ++j) {
        const int m = j + half * 8;
        const int s = s_samp[m];
        if (s >= 0) ocol[(size_t)s * FDIM] = fmaxf(acc[j], 0.0f);
    }
}

// ---------------- launcher ----------------

extern "C" void kernel_launch(void* const* d_in, const int* in_sizes, int n_in,
                              void* d_out, int out_size, void* d_ws, size_t ws_size,
                              hipStream_t stream) {
    const float* x    = (const float*)d_in[0];   // (8192, 256)
    const int*   idxs = (const int*)d_in[1];     // (8192,)
    const float* w    = (const float*)d_in[2];   // (64, 256, 256)
    const float* bias = (const float*)d_in[3];   // (64, 256)
    float* out = (float*)d_out;                  // (8192, 256)

    const int batch = in_sizes[1];

    // workspace layout: counts[64] | offsets[64] | cursor[64] | perm[batch]
    int* counts  = (int*)d_ws;
    int* offsets = counts + N_MODELS;
    int* cursor  = offsets + N_MODELS;
    int* perm    = cursor + N_MODELS;

    zero_counts_kernel<<<1, N_MODELS, 0, stream>>>(counts);
    count_kernel<<<(batch + 255) / 256, 256, 0, stream>>>(idxs, counts, batch);
    scan_kernel<<<1, 1, 0, stream>>>(counts, offsets, cursor);
    scatter_kernel<<<(batch + 255) / 256, 256, 0, stream>>>(idxs, cursor, perm, batch);

    const int maxTiles = (batch + TILE_M - 1) / TILE_M;   // worst case: all samples in one model
    dim3 grid(maxTiles, N_MODELS);
    moe_wmma_gemm_kernel<<<grid, 512, 0, stream>>>(x, w, bias, counts, offsets, perm, out);
}